// CoocGNN_57329223467073
// MI455X (gfx1250) — compile-verified
//
#include <hip/hip_runtime.h>
#include <hip/hip_bf16.h>

// ---------------------------------------------------------------------------
// CoocGNN forward pipeline for gfx1250 (MI455X), fp32 WMMA 16x16x4 everywhere.
// Outputs (flat, in order): W_adj[80,80], aw[80,2048], dlog[80],
// cam_vis[32,80,32,32], refined[32,80], total[1]
// ---------------------------------------------------------------------------

typedef __attribute__((ext_vector_type(2))) float v2f;
typedef __attribute__((ext_vector_type(8))) float v8f;

#define CC   80
#define BB   32
#define FC   2048
#define HWN  1024
#define HIDD 256
#define EDDD 64
#define TEMPI (1.0f/2.5f)

__device__ __forceinline__ float sigmoidf(float x) { return 1.0f / (1.0f + expf(-x)); }
__device__ __forceinline__ float bce_logits(float h, float t) {
    return fmaxf(h, 0.0f) - h * t + log1pf(expf(-fabsf(h)));
}
__device__ __forceinline__ v8f wmma4(v2f a, v2f b, v8f c) {
    return __builtin_amdgcn_wmma_f32_16x16x4_f32(false, a, false, b, (short)0, c, false, false);
}

// ---------------- generic fp32 WMMA GEMM: D = act(A[M,K]@B[K,N] + bias) -----
// grid = (N/16, M/16), block = 32 (one wave). K % 4 == 0, M,N % 16 == 0.
__global__ void k_wmma_gemm(const float* __restrict__ A, int lda,
                            const float* __restrict__ Bm, int ldb,
                            const float* __restrict__ bias,
                            float* __restrict__ D, int ldd,
                            int K, int act) {
    const int lane = threadIdx.x & 31;
    const int half = lane >> 4;
    const int l16  = lane & 15;
    const int m0 = blockIdx.y << 4;
    const int n0 = blockIdx.x << 4;
    const float* Arow = A + (long)(m0 + l16) * lda;
    const float* Bcol = Bm + (n0 + l16);
    v8f acc = {};
    for (int k = 0; k < K; k += 4) {
        const int ka = k + 2 * half;
        v2f a, b;
        a.x = Arow[ka];
        a.y = Arow[ka + 1];
        b.x = Bcol[(long)ka * ldb];
        b.y = Bcol[(long)(ka + 1) * ldb];
        acc = wmma4(a, b, acc);
    }
    const int col = n0 + l16;
    const float bv = bias ? bias[col] : 0.0f;
#pragma unroll
    for (int v = 0; v < 8; ++v) {
        const int row = m0 + v + half * 8;
        float val = acc[v] + bv;
        if (act) val = fmaxf(val, 0.0f);
        D[(long)row * ldd + col] = val;
    }
}

// ---------------- CAM: cam[b,c,n] = relu(sum_f aw[c,f] * feats[b,f,n]) ------
// Dominant GEMM: per wave computes a 16x64 output strip (4 accumulators),
// reusing the A fragment 4x per K-step to raise wmma:vmem ratio.
// grid = (HWN/64, CC/16, BB), block = 32
__global__ void k_wmma_cam(const float* __restrict__ aw,
                           const float* __restrict__ feats,
                           float* __restrict__ cam) {
    const int lane = threadIdx.x & 31;
    const int half = lane >> 4;
    const int l16  = lane & 15;
    const int m0 = blockIdx.y << 4;
    const int n0 = blockIdx.x << 6;   // 64 columns per wave
    const long b  = blockIdx.z;
    const float* Arow = aw + (long)(m0 + l16) * FC;
    const float* Bcol = feats + b * (long)FC * HWN + (n0 + l16);
    v8f acc0 = {}, acc1 = {}, acc2 = {}, acc3 = {};
#pragma unroll 2
    for (int k = 0; k < FC; k += 4) {
        const int ka = k + 2 * half;
        // speculative prefetch of the B operand two K-tiles ahead (gfx1250
        // global_prefetch_b8); feats streams once from HBM.
        if (k + 8 < FC)
            __builtin_prefetch(Bcol + (long)(ka + 8) * HWN, 0, 0);
        v2f a;
        a.x = Arow[ka];
        a.y = Arow[ka + 1];
        const float* Bk0 = Bcol + (long)ka * HWN;
        const float* Bk1 = Bcol + (long)(ka + 1) * HWN;
        v2f b0, b1, b2, b3;
        b0.x = Bk0[0];  b0.y = Bk1[0];
        b1.x = Bk0[16]; b1.y = Bk1[16];
        b2.x = Bk0[32]; b2.y = Bk1[32];
        b3.x = Bk0[48]; b3.y = Bk1[48];
        acc0 = wmma4(a, b0, acc0);
        acc1 = wmma4(a, b1, acc1);
        acc2 = wmma4(a, b2, acc2);
        acc3 = wmma4(a, b3, acc3);
    }
    float* Dp = cam + b * (long)CC * HWN;
    const int col = n0 + l16;
#pragma unroll
    for (int v = 0; v < 8; ++v) {
        const long row = m0 + v + half * 8;
        Dp[row * HWN + col]      = fmaxf(acc0[v], 0.0f);
        Dp[row * HWN + col + 16] = fmaxf(acc1[v], 0.0f);
        Dp[row * HWN + col + 32] = fmaxf(acc2[v], 0.0f);
        Dp[row * HWN + col + 48] = fmaxf(acc3[v], 0.0f);
    }
}

// ---------------- edge MLP: 16 edges per block, LDS-staged features ---------
// features[e,0:64]=Zd[i], [64:128]=Zd[j], [128:192]=|zi-zj|, [192:256]=zi*zj,
// [256]=cos[i,j], [257]=freq[i], [258]=freq[j], [259]=0 pad. grid=400, block=32
__global__ void k_edge_mlp(const float* __restrict__ Zd,
                           const float* __restrict__ cosm,
                           const float* __restrict__ freq,
                           const float* __restrict__ prior_pmi,
                           const float* __restrict__ em_w1, const float* __restrict__ em_b1,
                           const float* __restrict__ em_w2, const float* __restrict__ em_b2,
                           const float* __restrict__ em_w3, const float* __restrict__ em_b3,
                           float* __restrict__ r_out, float* __restrict__ h_out) {
    __shared__ float Fs[16][260];
    __shared__ float E1[16][128];
    __shared__ float E2[16][64];
    const int lane = threadIdx.x & 31;
    const int half = lane >> 4;
    const int l16  = lane & 15;
    const int n0 = blockIdx.x * 16;

    for (int idx = lane; idx < 16 * 260; idx += 32) {
        const int e  = idx / 260;
        const int kk = idx - e * 260;
        const int n  = n0 + e;
        const int i  = n / CC;
        const int j  = n - i * CC;
        float val;
        if (kk < 64)       val = Zd[i * EDDD + kk];
        else if (kk < 128) val = Zd[j * EDDD + (kk - 64)];
        else if (kk < 192) val = fabsf(Zd[i * EDDD + (kk - 128)] - Zd[j * EDDD + (kk - 128)]);
        else if (kk < 256) val = Zd[i * EDDD + (kk - 192)] * Zd[j * EDDD + (kk - 192)];
        else if (kk == 256) val = cosm[i * CC + j];
        else if (kk == 257) val = freq[i];
        else if (kk == 258) val = freq[j];
        else                val = 0.0f;  // pad K=259 -> 260
        Fs[e][kk] = val;
    }
    __syncthreads();

    // layer 1: [16,260] @ em_w1[259,128] (padded) -> relu -> E1[16,128]
    for (int nt = 0; nt < 8; ++nt) {
        const int col = nt * 16 + l16;
        v8f acc = {};
        for (int k = 0; k < 260; k += 4) {
            const int ka = k + 2 * half;
            v2f a, b;
            a.x = Fs[l16][ka];
            a.y = Fs[l16][ka + 1];
            b.x = (ka     < 259) ? em_w1[ka * 128 + col]       : 0.0f;
            b.y = (ka + 1 < 259) ? em_w1[(ka + 1) * 128 + col] : 0.0f;
            acc = wmma4(a, b, acc);
        }
        const float bv = em_b1[col];
#pragma unroll
        for (int v = 0; v < 8; ++v)
            E1[v + half * 8][col] = fmaxf(acc[v] + bv, 0.0f);
    }
    __syncthreads();

    // layer 2: [16,128] @ em_w2[128,64] -> relu -> E2[16,64]
    for (int nt = 0; nt < 4; ++nt) {
        const int col = nt * 16 + l16;
        v8f acc = {};
        for (int k = 0; k < 128; k += 4) {
            const int ka = k + 2 * half;
            v2f a, b;
            a.x = E1[l16][ka];
            a.y = E1[l16][ka + 1];
            b.x = em_w2[ka * 64 + col];
            b.y = em_w2[(ka + 1) * 64 + col];
            acc = wmma4(a, b, acc);
        }
        const float bv = em_b2[col];
#pragma unroll
        for (int v = 0; v < 8; ++v)
            E2[v + half * 8][col] = fmaxf(acc[v] + bv, 0.0f);
    }
    __syncthreads();

    // layer 3: [16,64] @ em_w3[64,1] ; h = pmi/T + r
    if (lane < 16) {
        const int n = n0 + lane;
        float rr = em_b3[0];
#pragma unroll 8
        for (int q = 0; q < 64; ++q) rr += E2[lane][q] * em_w3[q];
        r_out[n] = rr;
        h_out[n] = prior_pmi[n] * TEMPI + rr;
    }
}

// ---------------- elementwise / reduction helpers ---------------------------
__global__ void k_probs(const float* __restrict__ lg, float* __restrict__ probs) {
    const int i = blockIdx.x * blockDim.x + threadIdx.x;
    if (i < BB * CC) probs[i] = sigmoidf(lg[i]);
}

__global__ void k_pooled(const float* __restrict__ feats, float* __restrict__ pooled) {
    const int i = blockIdx.x * blockDim.x + threadIdx.x;
    if (i >= BB * FC) return;
    const float* p = feats + (long)i * HWN;
    float s = 0.0f;
    for (int n = 0; n < HWN; ++n) s += p[n];
    pooled[i] = s * (1.0f / HWN);
}

__global__ void k_freq(const float* __restrict__ probs, const int* __restrict__ labels,
                       float* __restrict__ freq, float* __restrict__ wsum,
                       float* __restrict__ present) {
    const int c = blockIdx.x * blockDim.x + threadIdx.x;
    if (c >= CC) return;
    float s = 0.0f; float pres = 0.0f;
    for (int b = 0; b < BB; ++b) {
        s += probs[b * CC + c];
        if ((float)labels[b * CC + c] > 0.0f) pres = 1.0f;
    }
    freq[c] = s * (1.0f / BB);
    wsum[c] = fmaxf(s, 1e-6f);
    present[c] = pres;
}

__global__ void k_proto(const float* __restrict__ probs, const float* __restrict__ pooled,
                        const float* __restrict__ wsum, float* __restrict__ proto) {
    const int i = blockIdx.x * blockDim.x + threadIdx.x;
    if (i >= CC * FC) return;
    const int c = i / FC, f = i - c * FC;
    float s = 0.0f;
    for (int b = 0; b < BB; ++b) s += probs[b * CC + c] * pooled[b * FC + f];
    proto[i] = s / wsum[c];
}

__global__ void k_rnorm(const float* __restrict__ proto, float* __restrict__ rn) {
    const int c = blockIdx.x * blockDim.x + threadIdx.x;
    if (c >= CC) return;
    const float* p = proto + (long)c * FC;
    float s = 0.0f;
    for (int f = 0; f < FC; ++f) s += p[f] * p[f];
    rn[c] = fmaxf(sqrtf(s), 1e-6f);
}

__global__ void k_cos(const float* __restrict__ proto, const float* __restrict__ rn,
                      float* __restrict__ cosm) {
    const int e = blockIdx.x * blockDim.x + threadIdx.x;
    if (e >= CC * CC) return;
    const int i = e / CC, j = e - i * CC;
    const float* pi = proto + (long)i * FC;
    const float* pj = proto + (long)j * FC;
    float s = 0.0f;
    for (int f = 0; f < FC; ++f) s += pi[f] * pj[f];
    float c = s / (rn[i] * rn[j]);
    cosm[e] = fminf(fmaxf(c, -1.0f), 1.0f);
}

__global__ void k_wadj(const float* __restrict__ h, float* __restrict__ Wout) {
    const int e = blockIdx.x * blockDim.x + threadIdx.x;
    if (e >= CC * CC) return;
    const int i = e / CC, j = e - i * CC;
    float w = 0.5f * (sigmoidf(h[i * CC + j]) + sigmoidf(h[j * CC + i]));
    Wout[e] = (i == j) ? 0.0f : w;
}

__global__ void k_rownorm_adj(const float* __restrict__ W, float* __restrict__ A) {
    const int i = blockIdx.x * blockDim.x + threadIdx.x;
    if (i >= CC) return;
    float s = 0.0f;
    for (int j = 0; j < CC; ++j) s += W[i * CC + j];
    const float inv = 1.0f / fmaxf(s, 1e-6f);
    for (int j = 0; j < CC; ++j) A[i * CC + j] = W[i * CC + j] * inv;
}

__global__ void k_zupdate(const float* __restrict__ Hn, const float* __restrict__ M1,
                          float* __restrict__ Z) {
    const int i = blockIdx.x * blockDim.x + threadIdx.x;
    if (i < CC * HIDD) Z[i] = fmaxf(Hn[i] + M1[i], 0.0f);
}

// softplus + row-normalize aw in place; grid = 80 rows, block = 256
__global__ void k_awnorm(float* __restrict__ aw) {
    __shared__ float red[256];
    float* row = aw + (long)blockIdx.x * FC;
    float local = 0.0f;
    for (int f = threadIdx.x; f < FC; f += 256) {
        const float x = row[f];
        const float sp = fmaxf(x, 0.0f) + log1pf(expf(-fabsf(x)));
        row[f] = sp;
        local += sp;
    }
    red[threadIdx.x] = local;
    __syncthreads();
    for (int s = 128; s > 0; s >>= 1) {
        if (threadIdx.x < s) red[threadIdx.x] += red[threadIdx.x + s];
        __syncthreads();
    }
    const float inv = 1.0f / fmaxf(red[0], 1e-6f);
    for (int f = threadIdx.x; f < FC; f += 256) row[f] *= inv;
}

__global__ void k_dlog(const float* __restrict__ Z, const float* __restrict__ bw,
                       const float* __restrict__ bb, float* __restrict__ dlog) {
    const int c = blockIdx.x * blockDim.x + threadIdx.x;
    if (c >= CC) return;
    float s = bb[0];
    for (int k = 0; k < HIDD; ++k) s += Z[c * HIDD + k] * bw[k];
    dlog[c] = s;
}

__global__ void k_refined(const float* __restrict__ lg, const float* __restrict__ probs,
                          const int* __restrict__ labels, const float* __restrict__ W,
                          const float* __restrict__ dlog, float* __restrict__ out) {
    const int i = blockIdx.x * blockDim.x + threadIdx.x;
    if (i >= BB * CC) return;
    const int b = i / CC, d = i - b * CC;
    const float yd = ((float)labels[b * CC + d] > 0.0f) ? 1.0f : 0.0f;
    float pos = 0.0f, psum = 0.0f;
    for (int c = 0; c < CC; ++c) {
        const float p = probs[b * CC + c];
        psum += p;
        const float yc = ((float)labels[b * CC + c] > 0.0f) ? 1.0f : 0.0f;
        pos += p * W[c * CC + d] * yc;
    }
    pos *= yd;
    const float neg = psum - pos;
    out[i] = lg[i] + 0.5f * pos - 0.25f * neg + dlog[d];
}

// cam min/max normalize in place; grid = (CC, BB), block = 256
__global__ void k_camnorm(float* __restrict__ cam) {
    __shared__ float smn[256], smx[256];
    float* row = cam + ((long)blockIdx.y * CC + blockIdx.x) * HWN;
    float mn = 1e30f, mx = -1e30f;
    for (int n = threadIdx.x; n < HWN; n += 256) {
        const float v = row[n];
        mn = fminf(mn, v);
        mx = fmaxf(mx, v);
    }
    smn[threadIdx.x] = mn; smx[threadIdx.x] = mx;
    __syncthreads();
    for (int s = 128; s > 0; s >>= 1) {
        if (threadIdx.x < s) {
            smn[threadIdx.x] = fminf(smn[threadIdx.x], smn[threadIdx.x + s]);
            smx[threadIdx.x] = fmaxf(smx[threadIdx.x], smx[threadIdx.x + s]);
        }
        __syncthreads();
    }
    const float m0 = smn[0];
    const float inv = 1.0f / (smx[0] - m0 + 1e-6f);
    for (int n = threadIdx.x; n < HWN; n += 256) row[n] = (row[n] - m0) * inv;
}

__global__ void k_cls_loss(const float* __restrict__ refined, const int* __restrict__ labels,
                           float* __restrict__ cls_out) {
    __shared__ float sl[256], sm[256];
    float ls = 0.0f, ms = 0.0f;
    for (int i = threadIdx.x; i < BB * CC; i += 256) {
        const float tgt = (float)labels[i];
        const float m = (tgt != -1.0f) ? 1.0f : 0.0f;
        const float st = (m > 0.0f) ? tgt : 0.0f;
        ls += bce_logits(refined[i], st) * m;
        ms += m;
    }
    sl[threadIdx.x] = ls; sm[threadIdx.x] = ms;
    __syncthreads();
    for (int s = 128; s > 0; s >>= 1) {
        if (threadIdx.x < s) {
            sl[threadIdx.x] += sl[threadIdx.x + s];
            sm[threadIdx.x] += sm[threadIdx.x + s];
        }
        __syncthreads();
    }
    if (threadIdx.x == 0) cls_out[0] = sl[0] / fmaxf(sm[0], 1.0f);
}

__global__ void k_total(const float* __restrict__ prior_pmi, const float* __restrict__ h,
                        const float* __restrict__ r, const float* __restrict__ W,
                        const float* __restrict__ present, const float* __restrict__ cls,
                        float* __restrict__ total_out) {
    __shared__ float s0[256], s1[256], s2[256], s3[256];
    // phase 1: n_pos / n_neg over masked edges
    float np = 0.0f, nn = 0.0f;
    for (int e = threadIdx.x; e < CC * CC; e += 256) {
        const int i = e / CC, j = e - i * CC;
        const float emf = (i != j && present[i] > 0.0f && present[j] > 0.0f) ? 1.0f : 0.0f;
        const float t = sigmoidf(prior_pmi[e] * TEMPI);
        if (t > 0.5f) np += emf; else nn += emf;
    }
    s0[threadIdx.x] = np; s1[threadIdx.x] = nn;
    __syncthreads();
    for (int s = 128; s > 0; s >>= 1) {
        if (threadIdx.x < s) { s0[threadIdx.x] += s0[threadIdx.x + s]; s1[threadIdx.x] += s1[threadIdx.x + s]; }
        __syncthreads();
    }
    const float n_pos = fmaxf(s0[0], 1.0f);
    const float n_neg = fmaxf(s1[0], 1.0f);
    const float w_pos = fminf(fmaxf(n_neg / n_pos, 1.0f), 10.0f);
    __syncthreads();
    // phase 2: edge loss, r_reg, n_edges, sparsity
    float el = 0.0f, rr = 0.0f, ne = 0.0f, wsum = 0.0f;
    for (int e = threadIdx.x; e < CC * CC; e += 256) {
        const int i = e / CC, j = e - i * CC;
        const float emf = (i != j && present[i] > 0.0f && present[j] > 0.0f) ? 1.0f : 0.0f;
        const float t = sigmoidf(prior_pmi[e] * TEMPI);
        const float wgt = (t > 0.5f) ? w_pos : 1.0f;
        el += bce_logits(h[e], t) * wgt * emf;
        rr += fabsf(r[e]) * emf;
        ne += emf;
        wsum += W[e];
    }
    s0[threadIdx.x] = el; s1[threadIdx.x] = rr; s2[threadIdx.x] = ne; s3[threadIdx.x] = wsum;
    __syncthreads();
    for (int s = 128; s > 0; s >>= 1) {
        if (threadIdx.x < s) {
            s0[threadIdx.x] += s0[threadIdx.x + s];
            s1[threadIdx.x] += s1[threadIdx.x + s];
            s2[threadIdx.x] += s2[threadIdx.x + s];
            s3[threadIdx.x] += s3[threadIdx.x + s];
        }
        __syncthreads();
    }
    if (threadIdx.x == 0) {
        const float n_edges = fmaxf(s2[0], 1.0f);
        const float edge_loss = s0[0] / n_edges;
        const float r_reg = 0.001f * s1[0] / n_edges;
        const float sparsity = s3[0] / (float)(CC * CC);
        total_out[0] = cls[0] + 0.1f * edge_loss + r_reg + 0.01f * sparsity;
    }
}

// ---------------------------------------------------------------------------
extern "C" void kernel_launch(void* const* d_in, const int* in_sizes, int n_in,
                              void* d_out, int out_size, void* d_ws, size_t ws_size,
                              hipStream_t stream) {
    const float* feats      = (const float*)d_in[0];
    const float* cls_logits = (const float*)d_in[1];
    const int*   img_labels = (const int*)d_in[2];
    const float* prior_pmi  = (const float*)d_in[3];
    const float* pp_w1 = (const float*)d_in[4];   const float* pp_b1 = (const float*)d_in[5];
    const float* pp_w2 = (const float*)d_in[6];   const float* pp_b2 = (const float*)d_in[7];
    const float* msg_w1 = (const float*)d_in[8];  const float* msg_b1 = (const float*)d_in[9];
    const float* msg_w2 = (const float*)d_in[10]; const float* msg_b2 = (const float*)d_in[11];
    const float* ed_w = (const float*)d_in[12];   const float* ed_b = (const float*)d_in[13];
    const float* em_w1 = (const float*)d_in[14];  const float* em_b1 = (const float*)d_in[15];
    const float* em_w2 = (const float*)d_in[16];  const float* em_b2 = (const float*)d_in[17];
    const float* em_w3 = (const float*)d_in[18];  const float* em_b3 = (const float*)d_in[19];
    const float* alpha_w = (const float*)d_in[20]; const float* alpha_b = (const float*)d_in[21];
    const float* bias_w = (const float*)d_in[22]; const float* bias_b = (const float*)d_in[23];

    // output layout
    float* out      = (float*)d_out;
    float* o_wadj   = out;                       // 6400
    float* o_aw     = out + 6400;                // 163840
    float* o_dlog   = out + 170240;              // 80
    float* o_cam    = out + 170320;              // 2621440
    float* o_ref    = out + 2791760;             // 2560
    float* o_total  = out + 2794320;             // 1

    // workspace layout (floats)
    float* w = (float*)d_ws;
    float* probs   = w;            w += BB * CC;     // 2560
    float* pooled  = w;            w += BB * FC;     // 65536
    float* proto   = w;            w += CC * FC;     // 163840
    float* rn      = w;            w += CC;
    float* cosm    = w;            w += CC * CC;
    float* freq    = w;            w += CC;
    float* wsum    = w;            w += CC;
    float* present = w;            w += CC;
    float* T1      = w;            w += CC * HIDD;   // scratch GEMM
    float* Hn      = w;            w += CC * HIDD;
    float* Zd      = w;            w += CC * EDDD;
    float* rbuf    = w;            w += CC * CC;
    float* hbuf    = w;            w += CC * CC;
    float* Amat    = w;            w += CC * CC;
    float* AZ      = w;            w += CC * HIDD;
    float* M1      = w;            w += CC * HIDD;
    float* Zb      = w;            w += CC * HIDD;
    float* clsws   = w;            w += 1;

    // 1. probs, pooled, freq/present, proto, norms, cos
    k_probs<<<(BB * CC + 255) / 256, 256, 0, stream>>>(cls_logits, probs);
    k_pooled<<<(BB * FC + 255) / 256, 256, 0, stream>>>(feats, pooled);
    k_freq<<<1, 128, 0, stream>>>(probs, img_labels, freq, wsum, present);
    k_proto<<<(CC * FC + 255) / 256, 256, 0, stream>>>(probs, pooled, wsum, proto);
    k_rnorm<<<1, 128, 0, stream>>>(proto, rn);
    k_cos<<<(CC * CC + 255) / 256, 256, 0, stream>>>(proto, rn, cosm);

    // 2. Hn = relu(proto@pp_w1+b1)@pp_w2+b2   (WMMA)
    k_wmma_gemm<<<dim3(HIDD / 16, CC / 16), 32, 0, stream>>>(proto, FC, pp_w1, HIDD, pp_b1, T1, HIDD, FC, 1);
    k_wmma_gemm<<<dim3(HIDD / 16, CC / 16), 32, 0, stream>>>(T1, HIDD, pp_w2, HIDD, pp_b2, Hn, HIDD, HIDD, 0);

    // 3. Zd = relu(Hn@ed_w+ed_b)              (WMMA)
    k_wmma_gemm<<<dim3(EDDD / 16, CC / 16), 32, 0, stream>>>(Hn, HIDD, ed_w, EDDD, ed_b, Zd, EDDD, HIDD, 1);

    // 4. edge MLP -> r, h                      (WMMA + LDS)
    k_edge_mlp<<<(CC * CC) / 16, 32, 0, stream>>>(Zd, cosm, freq, prior_pmi,
                                                  em_w1, em_b1, em_w2, em_b2, em_w3, em_b3,
                                                  rbuf, hbuf);

    // 5. W_adj (symmetrized, zero diag) and A (row-normalized)
    k_wadj<<<(CC * CC + 255) / 256, 256, 0, stream>>>(hbuf, o_wadj);
    k_rownorm_adj<<<1, 96, 0, stream>>>(o_wadj, Amat);

    // 6. one message-passing step: Z = relu(Hn + msg(A@Hn))   (WMMA x3)
    k_wmma_gemm<<<dim3(HIDD / 16, CC / 16), 32, 0, stream>>>(Amat, CC, Hn, HIDD, nullptr, AZ, HIDD, CC, 0);
    k_wmma_gemm<<<dim3(HIDD / 16, CC / 16), 32, 0, stream>>>(AZ, HIDD, msg_w1, HIDD, msg_b1, T1, HIDD, HIDD, 1);
    k_wmma_gemm<<<dim3(HIDD / 16, CC / 16), 32, 0, stream>>>(T1, HIDD, msg_w2, HIDD, msg_b2, M1, HIDD, HIDD, 0);
    k_zupdate<<<(CC * HIDD + 255) / 256, 256, 0, stream>>>(Hn, M1, Zb);

    // 7. heads: aw (WMMA + softplus rownorm), dlog
    k_wmma_gemm<<<dim3(FC / 16, CC / 16), 32, 0, stream>>>(Zb, HIDD, alpha_w, FC, alpha_b, o_aw, FC, HIDD, 0);
    k_awnorm<<<CC, 256, 0, stream>>>(o_aw);
    k_dlog<<<1, 96, 0, stream>>>(Zb, bias_w, bias_b, o_dlog);

    // 8. refined logits
    k_refined<<<(BB * CC + 255) / 256, 256, 0, stream>>>(cls_logits, probs, img_labels, o_wadj, o_dlog, o_ref);

    // 9. CAM (dominant GEMM, WMMA 16x64 strips) + min/max normalize in place
    k_wmma_cam<<<dim3(HWN / 64, CC / 16, BB), 32, 0, stream>>>(o_aw, feats, o_cam);
    k_camnorm<<<dim3(CC, BB), 256, 0, stream>>>(o_cam);

    // 10. losses -> total
    k_cls_loss<<<1, 256, 0, stream>>>(o_ref, img_labels, clsws);
    k_total<<<1, 256, 0, stream>>>(prior_pmi, hbuf, rbuf, o_wadj, present, clsws, o_total);

    (void)in_sizes; (void)n_in; (void)out_size; (void)ws_size;
}